// Net_36386962932143
// MI455X (gfx1250) — compile-verified
//
#include <hip/hip_runtime.h>
#include <math.h>

typedef float v2f __attribute__((ext_vector_type(2)));
typedef float v8f __attribute__((ext_vector_type(8)));

#define HID 128   // feature width of hidden layers (D == H == 128)

// ---------------------------------------------------------------------------
// Degree / normalization kernels
// ---------------------------------------------------------------------------
__global__ void k_deg_init(float* __restrict__ deg, int N) {
    int i = blockIdx.x * blockDim.x + threadIdx.x;
    if (i < N) deg[i] = 1.0f;                       // self-loop contributes 1
}

__global__ void k_deg_accum(const int* __restrict__ dst, float* __restrict__ deg, int E) {
    int e = blockIdx.x * blockDim.x + threadIdx.x;
    if (e < E) atomicAdd(&deg[dst[e]], 1.0f);
}

__global__ void k_dinv(float* __restrict__ d, int N) {
    int i = blockIdx.x * blockDim.x + threadIdx.x;
    if (i < N) d[i] = rsqrtf(fmaxf(d[i], 1.0f));
}

__global__ void k_zero(float* __restrict__ p, int n) {
    int i = blockIdx.x * blockDim.x + threadIdx.x;
    if (i < n) p[i] = 0.0f;
}

// ---------------------------------------------------------------------------
// WMMA fp32 GEMM:  C[N, ncols] = A[N, K] @ W[K, ncols] + bias
// One wave computes a 16 x (16*NT) strip: NT independent 16x16 accumulators.
// K is stepped by 8: per iteration, 2 A fragments + 2*NT B fragments are all
// loaded first (one base pointer, pure immediate offsets on the fast path),
// then 2*NT V_WMMA_F32_16X16X4_F32 are issued -> staggered s_wait_loadcnt,
// load latency hidden behind the independent WMMA chains.
//   grid.x = N/16 row tiles, grid.y = ncols / (16*NT) strip columns.
// MASKED=true (classifier, ncols=40, NT=3, grid.y=1): clamped addresses +
//   v_cndmask for loads (no EXEC change before WMMA), masked stores after.
// ---------------------------------------------------------------------------
template <int NT, bool MASKED>
__global__ void k_gemm_wmma(const float* __restrict__ A,
                            const float* __restrict__ W,
                            const float* __restrict__ bias,
                            float* __restrict__ C,
                            int K, int ncols) {
    const int rowBase = blockIdx.x * 16;
    const int colBase = blockIdx.y * (16 * NT);
    const int lane    = threadIdx.x;          // 0..31 (wave32)
    const int m       = lane & 15;            // M index (A) / N index (B, C)
    const int khalf   = (lane >> 4) << 1;     // 0 for lanes 0-15, 2 for 16-31

    const float* ap = A + (size_t)(rowBase + m) * K + khalf;  // 8B aligned

    v8f acc[NT];
    #pragma unroll
    for (int t = 0; t < NT; ++t) acc[t] = (v8f){};

    if (!MASKED) {
        // Fast path: single per-lane B base, every load at an imm offset.
        const float* wb = W + (size_t)khalf * ncols + colBase + m;
        for (int k0 = 0; k0 < K; k0 += 8) {
            // ---- phase 1: load everything for this K-superstep ----
            v2f a0 = *(const v2f*)(ap + k0);          // global_load_b64
            v2f a1 = *(const v2f*)(ap + k0 + 4);      // imm offset +16B
            const float* wr = wb + (size_t)k0 * ncols;
            v2f b0[NT], b1[NT];
            #pragma unroll
            for (int t = 0; t < NT; ++t) {
                b0[t].x = wr[16 * t];                 // +{0,64,128,192}B
                b0[t].y = wr[(size_t)ncols + 16 * t]; // +512B
                b1[t].x = wr[(size_t)4 * ncols + 16 * t]; // +2048B
                b1[t].y = wr[(size_t)5 * ncols + 16 * t]; // +2560B
            }
            // ---- phase 2: 2*NT WMMAs in NT independent chains ----
            #pragma unroll
            for (int t = 0; t < NT; ++t)
                acc[t] = __builtin_amdgcn_wmma_f32_16x16x4_f32(
                    false, a0, false, b0[t], (short)0, acc[t], false, false);
            #pragma unroll
            for (int t = 0; t < NT; ++t)
                acc[t] = __builtin_amdgcn_wmma_f32_16x16x4_f32(
                    false, a1, false, b1[t], (short)0, acc[t], false, false);
        }
        const int mb = (lane >> 4) << 3;      // rows 0..7 or 8..15
        #pragma unroll
        for (int t = 0; t < NT; ++t) {
            float bv = bias[colBase + 16 * t + m];
            #pragma unroll
            for (int r = 0; r < 8; ++r) {
                int row = rowBase + mb + r;
                C[(size_t)row * ncols + colBase + 16 * t + m] = acc[t][r] + bv;
            }
        }
    } else {
        // Masked path (column overhang): clamped addresses + selects.
        int col[NT]; bool ok[NT]; int colc[NT];
        #pragma unroll
        for (int t = 0; t < NT; ++t) {
            col[t]  = colBase + 16 * t + m;
            ok[t]   = (col[t] < ncols);
            colc[t] = ok[t] ? col[t] : 0;
        }
        const float* wp = W + (size_t)khalf * ncols;
        for (int k0 = 0; k0 < K; k0 += 8) {
            v2f a0 = *(const v2f*)(ap + k0);
            v2f a1 = *(const v2f*)(ap + k0 + 4);
            const float* wr = wp + (size_t)k0 * ncols;
            v2f b0[NT], b1[NT];
            #pragma unroll
            for (int t = 0; t < NT; ++t) {
                float v00 = wr[colc[t]];
                float v01 = wr[(size_t)ncols + colc[t]];
                float v10 = wr[(size_t)4 * ncols + colc[t]];
                float v11 = wr[(size_t)5 * ncols + colc[t]];
                b0[t].x = ok[t] ? v00 : 0.0f;  // v_cndmask, no EXEC change
                b0[t].y = ok[t] ? v01 : 0.0f;
                b1[t].x = ok[t] ? v10 : 0.0f;
                b1[t].y = ok[t] ? v11 : 0.0f;
            }
            #pragma unroll
            for (int t = 0; t < NT; ++t)
                acc[t] = __builtin_amdgcn_wmma_f32_16x16x4_f32(
                    false, a0, false, b0[t], (short)0, acc[t], false, false);
            #pragma unroll
            for (int t = 0; t < NT; ++t)
                acc[t] = __builtin_amdgcn_wmma_f32_16x16x4_f32(
                    false, a1, false, b1[t], (short)0, acc[t], false, false);
        }
        const int mb = (lane >> 4) << 3;
        #pragma unroll
        for (int t = 0; t < NT; ++t) {
            if (ok[t]) {
                float bv = bias[colc[t]];
                #pragma unroll
                for (int r = 0; r < 8; ++r) {
                    int row = rowBase + mb + r;
                    C[(size_t)row * ncols + col[t]] = acc[t][r] + bv;
                }
            }
        }
    }
}

// ---------------------------------------------------------------------------
// Edge scatter: agg[dst] += h[src] * (dinv[src]*dinv[dst]).
// One wave per edge, 4 floats per lane (128 features / 32 lanes).
// ---------------------------------------------------------------------------
__global__ void k_scatter(const int* __restrict__ src, const int* __restrict__ dst,
                          const float* __restrict__ dinv,
                          const float* __restrict__ h,
                          float* __restrict__ agg, int E) {
    int gid  = blockIdx.x * blockDim.x + threadIdx.x;
    int e    = gid >> 5;
    int lane = gid & 31;
    if (e >= E) return;
    int s = src[e], d = dst[e];
    float nrm = dinv[s] * dinv[d];
    float4 hv = ((const float4*)(h + (size_t)s * HID))[lane];
    float* ap = agg + (size_t)d * HID + lane * 4;
    atomicAdd(ap + 0, hv.x * nrm);
    atomicAdd(ap + 1, hv.y * nrm);
    atomicAdd(ap + 2, hv.z * nrm);
    atomicAdd(ap + 3, hv.w * nrm);
}

// ---------------------------------------------------------------------------
// Self-loop contribution + ReLU: out = relu(agg + dinv[i]^2 * hlin)
// ---------------------------------------------------------------------------
__global__ void k_selfloop_relu(const float* __restrict__ agg,
                                const float* __restrict__ hlin,
                                const float* __restrict__ dinv,
                                float* __restrict__ out, int total) {
    int idx = blockIdx.x * blockDim.x + threadIdx.x;
    if (idx >= total) return;
    int i = idx >> 7;                         // / 128
    float di = dinv[i];
    float v  = agg[idx] + di * di * hlin[idx];
    out[idx] = fmaxf(v, 0.0f);
}

// ---------------------------------------------------------------------------
// Row-wise L2 normalize in place (one wave per row, 4 floats/lane).
// ---------------------------------------------------------------------------
__global__ void k_l2norm(float* __restrict__ h, int N) {
    int row  = (blockIdx.x * blockDim.x + threadIdx.x) >> 5;
    int lane = threadIdx.x & 31;
    if (row >= N) return;
    float4* p = ((float4*)(h + (size_t)row * HID)) + lane;
    float4 v = *p;
    float ss = v.x * v.x + v.y * v.y + v.z * v.z + v.w * v.w;
    #pragma unroll
    for (int off = 16; off > 0; off >>= 1)
        ss += __shfl_xor(ss, off, 32);
    float scale = 1.0f / fmaxf(sqrtf(ss), 1e-12f);
    v.x *= scale; v.y *= scale; v.z *= scale; v.w *= scale;
    *p = v;
}

// ---------------------------------------------------------------------------
// Host-side launcher
// ---------------------------------------------------------------------------
extern "C" void kernel_launch(void* const* d_in, const int* in_sizes, int n_in,
                              void* d_out, int out_size, void* d_ws, size_t ws_size,
                              hipStream_t stream) {
    const float* x  = (const float*)d_in[0];
    const int*   ei = (const int*)d_in[1];
    const float* W1 = (const float*)d_in[2];
    const float* b1 = (const float*)d_in[3];
    const float* W2 = (const float*)d_in[4];
    const float* b2 = (const float*)d_in[5];
    const float* Wc = (const float*)d_in[6];
    const float* bc = (const float*)d_in[7];
    float* out = (float*)d_out;

    const int N = in_sizes[0] / HID;          // 10000
    const int E = in_sizes[1] / 2;            // 640000
    const int C = in_sizes[7];                // 40
    const int* src = ei;
    const int* dst = ei + E;

    float* ws   = (float*)d_ws;
    float* dinv = ws;                         // [N]
    float* bufA = dinv + N;                   // [N*HID]
    float* bufB = bufA + (size_t)N * HID;     // [N*HID]

    const int total = N * HID;
    const int rowT  = N / 16;                 // 625 row tiles
    const dim3 waveBlk(32, 1, 1);

    // --- degree -> dinv ---
    k_deg_init<<<(N + 255) / 256, 256, 0, stream>>>(dinv, N);
    k_deg_accum<<<(E + 255) / 256, 256, 0, stream>>>(dst, dinv, E);
    k_dinv<<<(N + 255) / 256, 256, 0, stream>>>(dinv, N);

    // --- layer 1: linear (WMMA) -> propagate -> relu ---
    k_gemm_wmma<4, false><<<dim3(rowT, HID / 64), waveBlk, 0, stream>>>(x, W1, b1, bufA, HID, HID);
    k_zero<<<(total + 255) / 256, 256, 0, stream>>>(bufB, total);
    {
        long long thr = (long long)E * 32;
        k_scatter<<<(unsigned)((thr + 255) / 256), 256, 0, stream>>>(src, dst, dinv, bufA, bufB, E);
    }
    k_selfloop_relu<<<(total + 255) / 256, 256, 0, stream>>>(bufB, bufA, dinv, bufA, total);

    // --- layer 2: linear (WMMA) -> propagate -> relu ---
    k_gemm_wmma<4, false><<<dim3(rowT, HID / 64), waveBlk, 0, stream>>>(bufA, W2, b2, bufB, HID, HID);
    k_zero<<<(total + 255) / 256, 256, 0, stream>>>(bufA, total);
    {
        long long thr = (long long)E * 32;
        k_scatter<<<(unsigned)((thr + 255) / 256), 256, 0, stream>>>(src, dst, dinv, bufB, bufA, E);
    }
    k_selfloop_relu<<<(total + 255) / 256, 256, 0, stream>>>(bufA, bufB, dinv, bufB, total);

    // --- L2 normalize rows, then classifier (WMMA, 40 cols -> 3 tiles) ---
    k_l2norm<<<(N * 32 + 255) / 256, 256, 0, stream>>>(bufB, N);
    k_gemm_wmma<3, true><<<dim3(rowT, 1), waveBlk, 0, stream>>>(bufB, Wc, bc, out, HID, C);
}